// Voxelization_40785009443381
// MI455X (gfx1250) — compile-verified
//
#include <hip/hip_runtime.h>
#include <stdint.h>

// ---------------- problem constants ----------------
#define NXv      200
#define NYv      200
#define NZv      12
#define SENTv    (NXv * NYv * NZv)      // 480000
#define MAXVOX   500000
#define MAXPTS   32
#define CAPv     32                     // per-voxel index-list capacity
static_assert(CAPv == MAXPTS, "kept-count logic assumes CAP == MAX_PTS");

#define SCAN_BLOCK 256
#define SCAN_ITEMS 8
#define SCAN_TILE  (SCAN_BLOCK * SCAN_ITEMS)                 // 2048
#define NSCANBLK   ((SENTv + SCAN_TILE - 1) / SCAN_TILE)     // 235
static_assert(NSCANBLK <= SCAN_BLOCK, "scan2 fits in one block");

// ---------------- workspace layout (bytes) ----------------
#define WS_OFF_CNT    ((size_t)0)
#define WS_OFF_DENSE  ((size_t)1920000)              // SENT*4
#define WS_OFF_BLK    ((size_t)3840000)
#define WS_OFF_LISTS  ((size_t)3841024)              // 128B aligned
// total needed: 3,841,024 + 480000*32*4 = 65,281,024 bytes

// ---------------- CDNA5 async global->LDS path (guarded) ----------------
#if defined(__HIP_DEVICE_COMPILE__) && defined(__gfx1250__) && \
    __has_builtin(__builtin_amdgcn_global_load_async_to_lds_b128)
#define HAVE_ASYNC_LDS 1
// Builtin prototype (from hipcc diagnostic): param 1 is
//   'int __attribute__((vector_size(16))) __device__ *'  (global / AS1),
// param 2 is the LDS-side pointer (AS3), then two integer immediates.
typedef int v4i_ __attribute__((vector_size(16)));
typedef __attribute__((address_space(1))) v4i_ gv4i;
typedef __attribute__((address_space(3))) v4i_ lv4i;
#if __has_builtin(__builtin_amdgcn_s_wait_asynccnt)
#define ASYNC_WAIT0() __builtin_amdgcn_s_wait_asynccnt(0)
#else
#define ASYNC_WAIT0() asm volatile("s_wait_asynccnt 0" ::: "memory")
#endif
#else
#define HAVE_ASYNC_LDS 0
#endif

// Stage SCAN_TILE ints of the count array into LDS.
__device__ __forceinline__ void load_cnt_tile(const int* __restrict__ gsrc,
                                              int* tile) {
#if HAVE_ASYNC_LDS
  // 2048 ints = 8192 B; 256 lanes x 16 B x 2 iterations, DMA'd past VALU.
  #pragma unroll
  for (int it = 0; it < 2; ++it) {
    const int e = (it * SCAN_BLOCK + (int)threadIdx.x) * 4;  // int index, 16B/lane
    __builtin_amdgcn_global_load_async_to_lds_b128(
        (gv4i*)(gsrc + e), (lv4i*)(tile + e), 0, 0);
  }
  ASYNC_WAIT0();
  __syncthreads();
#else
  #pragma unroll
  for (int it = 0; it < SCAN_ITEMS; ++it)
    tile[it * SCAN_BLOCK + threadIdx.x] = gsrc[it * SCAN_BLOCK + threadIdx.x];
  __syncthreads();
#endif
}

// ---------------- kernels ----------------

// Default vox_center rows: ucoord==0 -> LB + VOX/2.
__global__ void fill_center_k(float* __restrict__ center) {
  int v = blockIdx.x * blockDim.x + threadIdx.x;
  if (v >= MAXVOX) return;
  float* c = center + (size_t)3 * v;
  c[0] = -49.75f; c[1] = -49.75f; c[2] = -2.75f;
}

// Per point: bin + range filter + ticketed insert into per-voxel list.
__global__ void points_k(const float* __restrict__ pc, int n,
                         int* __restrict__ cnt, int* __restrict__ lists) {
  int i = blockIdx.x * blockDim.x + threadIdx.x;
  if (i >= n) return;
  if (i + 2048 < n) __builtin_prefetch(pc + (size_t)3 * i + 3 * 2048, 0, 1);
  const float x = pc[3 * (size_t)i + 0];
  const float y = pc[3 * (size_t)i + 1];
  const float z = pc[3 * (size_t)i + 2];
  // inb uses pc vs LB/ub exactly like the reference (ub = 50,50,3 exact in f32)
  if (!(x >= -50.f && x < 50.f && y >= -50.f && y < 50.f &&
        z >= -3.f && z < 3.f))
    return;
  // (p - LB)/0.5 : the divide by 0.5 is an exact scale, identical to *2.0f
  const int cx = (int)floorf((x + 50.f) * 2.f);
  const int cy = (int)floorf((y + 50.f) * 2.f);
  const int cz = (int)floorf((z + 3.f) * 2.f);
  const int lin = (cx * NYv + cy) * NZv + cz;
  // boundary-rounding can push lin >= SENT; reference drops those from the
  // scatter (valid = lin < SENT) and they never perturb dense indices of
  // valid voxels (they sort after all of them).
  if (lin < 0 || lin >= SENTv) return;
  const int t = atomicAdd(&cnt[lin], 1);
  if (t < CAPv) lists[(size_t)lin * CAPv + t] = i;
}

// Scan pass 1: per-block occupancy sums.
__global__ void scan1_k(const int* __restrict__ cnt, int* __restrict__ blkSum) {
  __shared__ int tile[SCAN_TILE];
  __shared__ int red[SCAN_BLOCK];
  const int base = blockIdx.x * SCAN_TILE;
  load_cnt_tile(cnt + base, tile);
  const int t = threadIdx.x;
  int s = 0;
  #pragma unroll
  for (int j = 0; j < SCAN_ITEMS; ++j) {
    const int idx = base + t * SCAN_ITEMS + j;
    s += (idx < SENTv && tile[t * SCAN_ITEMS + j] > 0) ? 1 : 0;
  }
  red[t] = s;
  __syncthreads();
  for (int off = SCAN_BLOCK / 2; off > 0; off >>= 1) {
    if (t < off) red[t] += red[t + off];
    __syncthreads();
  }
  if (t == 0) blkSum[blockIdx.x] = red[0];
}

// Scan pass 2: exclusive scan of the block sums (single block).
__global__ void scan2_k(int* __restrict__ blkSum, int nblk) {
  __shared__ int sh[SCAN_BLOCK];
  const int t = threadIdx.x;
  const int v = (t < nblk) ? blkSum[t] : 0;
  sh[t] = v;
  __syncthreads();
  for (int off = 1; off < SCAN_BLOCK; off <<= 1) {
    const int add = (t >= off) ? sh[t - off] : 0;
    __syncthreads();
    sh[t] += add;
    __syncthreads();
  }
  if (t < nblk) blkSum[t] = sh[t] - v;  // exclusive
}

// Scan pass 3: per-element exclusive prefix -> dense voxel index.
__global__ void scan3_k(const int* __restrict__ cnt,
                        const int* __restrict__ blkOff,
                        int* __restrict__ dense) {
  __shared__ int tile[SCAN_TILE];
  __shared__ int sh[SCAN_BLOCK];
  const int base = blockIdx.x * SCAN_TILE;
  load_cnt_tile(cnt + base, tile);
  const int t = threadIdx.x;
  int f[SCAN_ITEMS];
  int s = 0;
  #pragma unroll
  for (int j = 0; j < SCAN_ITEMS; ++j) {
    const int idx = base + t * SCAN_ITEMS + j;
    f[j] = (idx < SENTv && tile[t * SCAN_ITEMS + j] > 0) ? 1 : 0;
    s += f[j];
  }
  sh[t] = s;
  __syncthreads();
  for (int off = 1; off < SCAN_BLOCK; off <<= 1) {
    const int add = (t >= off) ? sh[t - off] : 0;
    __syncthreads();
    sh[t] += add;
    __syncthreads();
  }
  int run = blkOff[blockIdx.x] + sh[t] - s;  // exclusive for this thread
  #pragma unroll
  for (int j = 0; j < SCAN_ITEMS; ++j) {
    const int idx = base + t * SCAN_ITEMS + j;
    if (idx < SENTv) dense[idx] = run;
    run += f[j];
  }
}

// Per occupied voxel: stable-order gather, centroid, decorated outputs.
__global__ void voxel_k(const float* __restrict__ pc,
                        const int* __restrict__ cnt,
                        const int* __restrict__ dense,
                        const int* __restrict__ lists,
                        float* __restrict__ out_voxel,
                        float* __restrict__ out_coord,
                        float* __restrict__ out_mask,
                        float* __restrict__ out_center) {
  const int lin = blockIdx.x * blockDim.x + threadIdx.x;
  if (lin >= SENTv) return;
  const int c = cnt[lin];
  if (c == 0) return;
  const int m = (c < CAPv) ? c : CAPv;          // kept == counts (CAP==MAX_PTS)
  const int v = dense[lin];
  const int* __restrict__ row = lists + (size_t)lin * CAPv;
  float* __restrict__ vrow = out_voxel + (size_t)v * (MAXPTS * 6);
  float* __restrict__ mrow = out_mask + (size_t)v * MAXPTS;

  // Traverse stored point indices in ascending original order (== reference's
  // stable sort) regardless of nondeterministic atomic ticket order.
  float sx = 0.f, sy = 0.f, sz = 0.f;
  int prev = -1;
  for (int k = 0; k < m; ++k) {
    int cur = 0x7fffffff;
    for (int t2 = 0; t2 < m; ++t2) {
      const int j = row[t2];
      cur = (j > prev && j < cur) ? j : cur;
    }
    prev = cur;
    const float px = pc[3 * (size_t)cur + 0];
    const float py = pc[3 * (size_t)cur + 1];
    const float pz = pc[3 * (size_t)cur + 2];
    vrow[k * 6 + 0] = px;
    vrow[k * 6 + 1] = py;
    vrow[k * 6 + 2] = pz;
    mrow[k] = 1.0f;
    sx += px; sy += py; sz += pz;   // slot order, matching jnp.sum
  }
  const float fm = (float)m;        // m >= 1
  const float ctx = sx / fm, cty = sy / fm, ctz = sz / fm;  // true divide
  for (int k = 0; k < m; ++k) {
    vrow[k * 6 + 3] = vrow[k * 6 + 0] - ctx;
    vrow[k * 6 + 4] = vrow[k * 6 + 1] - cty;
    vrow[k * 6 + 5] = vrow[k * 6 + 2] - ctz;
  }

  const int cz = lin % NZv;
  const int rem = lin / NZv;
  const int cy = rem % NYv;
  const int cx = rem / NYv;
  float* __restrict__ crow = out_coord + (size_t)v * 4;
  crow[0] = 0.f;
  crow[1] = (float)cx;
  crow[2] = (float)cy;
  crow[3] = (float)cz;
  float* __restrict__ ctr = out_center + (size_t)v * 3;
  ctr[0] = (float)cx * 0.5f + (-49.75f);
  ctr[1] = (float)cy * 0.5f + (-49.75f);
  ctr[2] = (float)cz * 0.5f + (-2.75f);
}

// ---------------- launch ----------------
extern "C" void kernel_launch(void* const* d_in, const int* in_sizes, int n_in,
                              void* d_out, int out_size, void* d_ws,
                              size_t ws_size, hipStream_t stream) {
  const float* pc = (const float*)d_in[0];
  const int n = in_sizes[0] / 3;

  float* out = (float*)d_out;
  float* out_voxel  = out;                                       // [V,32,6]
  float* out_coord  = out_voxel + (size_t)MAXVOX * MAXPTS * 6;   // [V,4]
  float* out_mask   = out_coord + (size_t)MAXVOX * 4;            // [V,32]
  float* out_center = out_mask  + (size_t)MAXVOX * MAXPTS;       // [V,3]

  char* ws = (char*)d_ws;
  int* cnt    = (int*)(ws + WS_OFF_CNT);
  int* dense  = (int*)(ws + WS_OFF_DENSE);
  int* blkSum = (int*)(ws + WS_OFF_BLK);
  int* lists  = (int*)(ws + WS_OFF_LISTS);

  // Zero voxel_out + coord_out + mask (456 MB) via memset node; centers filled
  // with their "ucoord==0" default by a kernel, then sparse overwrites.
  const size_t zero_floats = (size_t)MAXVOX * (MAXPTS * 6 + 4 + MAXPTS);
  (void)hipMemsetAsync(out, 0, zero_floats * sizeof(float), stream);
  (void)hipMemsetAsync(cnt, 0, (size_t)SENTv * sizeof(int), stream);

  fill_center_k<<<(MAXVOX + 255) / 256, 256, 0, stream>>>(out_center);
  points_k<<<(n + 255) / 256, 256, 0, stream>>>(pc, n, cnt, lists);
  scan1_k<<<NSCANBLK, SCAN_BLOCK, 0, stream>>>(cnt, blkSum);
  scan2_k<<<1, SCAN_BLOCK, 0, stream>>>(blkSum, NSCANBLK);
  scan3_k<<<NSCANBLK, SCAN_BLOCK, 0, stream>>>(cnt, blkSum, dense);
  voxel_k<<<(SENTv + 255) / 256, 256, 0, stream>>>(
      pc, cnt, dense, lists, out_voxel, out_coord, out_mask, out_center);
}